// RadonTransform_29961691857454
// MI455X (gfx1250) — compile-verified
//
#include <hip/hip_runtime.h>

// Radon transform: x[32,1,512,512] f32 -> sino[32,1,360,725] f32
// One workgroup per angle. lane = batch image (wave32 == 32 batches).
// Detector bin is wave-uniform -> conflict-free ds_add_f32 (32 consecutive banks).
// Axis choice is a static partition of angles: theta_a = a*pi/359,
//   |cos|<=|sin|  <=>  a in [90, 269]  -> x-major (templated, b128 loads)
//   otherwise                           -> y-major (4x b32 strided loads)
// Epilogue streams LDS -> global with global_store_async_from_lds_b32 (ASYNCcnt).

#define IMG 512
#define NA 360
#define ND 725
#define BCN 32
#define HW (IMG * IMG)
#define LDS_ROWS (ND + 1)  // row 725 = dump bucket for invalid pixels

template <bool XMAJOR>
__global__ __launch_bounds__(256) void radon_scatter_kernel(
    const float* __restrict__ x, float* __restrict__ out, int a_base) {
  __shared__ float sino[LDS_ROWS * BCN];  // 726*32*4 = 92,928 B (CDNA5 320KB LDS)

  const int a    = a_base + blockIdx.x;  // angle
  const int tid  = threadIdx.x;
  const int lane = tid & 31;             // batch index (B*C == 32 == wave32)
  const int wave = tid >> 5;             // 0..7

  // ---- zero accumulator ----
  for (int i = tid; i < LDS_ROWS * BCN; i += 256) sino[i] = 0.0f;
  __syncthreads();

  // ---- geometry (matches jnp.linspace exactly) ----
  const float theta = (float)a * (3.14159265358979323846f / 359.0f);
  float snt, cst;
  __sincosf(theta, &snt, &cst);
  const float S = 512.0f / 511.0f;  // coord spacing: linspace(-256,256,512)

  // proj = cx*cos + cy*sin. Fast axis = the one with the smaller coefficient
  // -> longest same-bin runs -> fewest LDS atomics.
  const float cf = XMAJOR ? cst : snt;      // fast-axis coefficient
  const float cs = XMAJOR ? snt : cst;      // slow-axis (per-line) coefficient
  constexpr int strideF = XMAJOR ? 1 : IMG; // element stride along fast axis
  constexpr int strideS = XMAJOR ? IMG : 1; // element stride between lines

  const float* xb = x + (size_t)lane * HW;  // this lane's image

  for (int l = wave; l < IMG; l += 8) {
    const float c_l   = fmaf((float)l, S, -256.0f);
    const float pbase = fmaf(c_l, cs, 362.0f);  // (proj + 362) partial
    const float* lp   = xb + (size_t)l * (size_t)strideS;
    if (l + 8 < IMG)
      __builtin_prefetch(xb + (size_t)(l + 8) * (size_t)strideS, 0, 1);  // global_prefetch_b8

    int   cur = -1;
    float acc = 0.0f;
    for (int i = 0; i < IMG; i += 4) {
      float vv[4];
      if constexpr (XMAJOR) {
        const float4 q = *(const float4*)(lp + i);  // global_load_b128: 4 pixels/lane
        vv[0] = q.x; vv[1] = q.y; vv[2] = q.z; vv[3] = q.w;
      } else {
        vv[0] = lp[(size_t)(i + 0) * IMG];          // 4 independent b32 loads,
        vv[1] = lp[(size_t)(i + 1) * IMG];          // issued back-to-back
        vv[2] = lp[(size_t)(i + 2) * IMG];
        vv[3] = lp[(size_t)(i + 3) * IMG];
      }
#pragma unroll
      for (int k = 0; k < 4; ++k) {
        const float cw = fmaf((float)(i + k), S, -256.0f);
        const float pj = fmaf(cw, cf, pbase);   // proj + 362
        const int   j  = __float2int_rn(pj);    // round-to-nearest-even
        // valid: j in [0,724] and proj in [det-0.5, det+0.5), det = j-362
        const int ok = (j >= 0) & (j <= ND - 1) &
                       (pj >= (float)j - 0.5f) & (pj < (float)j + 0.5f);
        const int b = ok ? j : ND;  // dump row for invalid pixels
        // b is wave-uniform -> uniform branch (s_cbranch), no divergence
        if (b != cur) {
          if (cur >= 0) atomicAdd(&sino[cur * BCN + lane], acc);  // ds_add_f32
          cur = b;
          acc = vv[k];
        } else {
          acc += vv[k];  // run-length compaction in a register
        }
      }
    }
    if (cur >= 0) atomicAdd(&sino[cur * BCN + lane], acc);
  }
  __syncthreads();

  // ---- epilogue: stream LDS -> global via async store (ASYNCcnt path) ----
  // out[bc*(NA*ND) + a*ND + d] = sino[d*BCN + bc]
  // Lanes take consecutive d -> coalesced 128B global bursts; no int division.
  for (int bc = wave; bc < BCN; bc += 8) {
    const float* obase = out + (size_t)bc * (size_t)(NA * ND) + (size_t)a * (size_t)ND;
    for (int d = lane; d < ND; d += 32) {
      const unsigned long long gaddr = (unsigned long long)(uintptr_t)(obase + d);
      const unsigned int loff =
          (unsigned int)(uintptr_t)(&sino[d * BCN + bc]);  // low 32 bits = LDS offset
      asm volatile("global_store_async_from_lds_b32 %0, %1, off"
                   :
                   : "v"(gaddr), "v"(loff)
                   : "memory");
    }
  }
  asm volatile("s_wait_asynccnt 0" ::: "memory");
}

extern "C" void kernel_launch(void* const* d_in, const int* in_sizes, int n_in,
                              void* d_out, int out_size, void* d_ws, size_t ws_size,
                              hipStream_t stream) {
  const float* x = (const float*)d_in[0];
  float* out     = (float*)d_out;
  (void)in_sizes; (void)n_in; (void)out_size; (void)d_ws; (void)ws_size;

  // Static angle partition by scan axis: theta_a = a*pi/359.
  //   a in [0,89]    : |cos| > |sin|  -> y-major
  //   a in [90,269]  : |cos| <= |sin| -> x-major
  //   a in [270,359] : |cos| > |sin|  -> y-major
  radon_scatter_kernel<false><<<dim3(90),  dim3(256), 0, stream>>>(x, out, 0);
  radon_scatter_kernel<true ><<<dim3(180), dim3(256), 0, stream>>>(x, out, 90);
  radon_scatter_kernel<false><<<dim3(90),  dim3(256), 0, stream>>>(x, out, 270);
}